// DetectionLayer_36103495090555
// MI455X (gfx1250) — compile-verified
//
#include <hip/hip_runtime.h>

// Detection decode: out[i, :] = (in[i,5] > 0.5) ? decode(in[i,:], an[i,:]) : 0
//   out[i,0] = in[i,0]*an[i,2] + an[i,0]
//   out[i,1] = in[i,1]*an[i,3] + an[i,1]
//   out[i,2] = exp(in[i,2])*an[i,2]
//   out[i,3] = exp(in[i,3])*an[i,3]
//   out[i,c] = in[i,c]                      (c >= 4)
//
// Memory-bound: ~712 MB read + ~680 MB write -> ~60us floor @ 23.3 TB/s.
// No GEMM -> no WMMA. gfx1250 levers: b128 VMEM, NT load/store hints,
// global_prefetch_b8, wave32 occupancy. The special-column decode is a single
// EXEC-masked branch per element with a branchless (cndmask) interior, so the
// hot loop avoids the nested-branch SALU blowup.

typedef float v4f __attribute__((ext_vector_type(4)));

constexpr unsigned int kC   = 85u;
constexpr float        kThr = 0.5f;

__global__ __launch_bounds__(256)
void detection_decode_kernel(const float* __restrict__ in,
                             const float* __restrict__ an,
                             float* __restrict__ out,
                             unsigned int n_elems,   // N * 85
                             unsigned int n_vec)     // ceil(n_elems / 4)
{
    const unsigned int base = blockIdx.x * (blockDim.x * 2u) + threadIdx.x;

#pragma unroll
    for (int it = 0; it < 2; ++it) {
        const unsigned int v = base + (unsigned int)it * 256u;
        if (v >= n_vec) return;                 // v grows with it -> safe

        const unsigned int e0 = v * 4u;

        // Stream the read set ~128 KB ahead into GL2 (speculative; dropped on
        // translation miss). Lowers to gfx1250 global_prefetch_b8.
        __builtin_prefetch(in + e0 + 32768u, 0, 0);

        // Recover (row, col) of the first element; one mulhi-divide by 85.
        const unsigned int row0  = e0 / kC;
        const unsigned int c0    = e0 - row0 * kC;
        const bool         cross = (c0 + 3u) >= kC;      // vector spans 2 rows
        const unsigned int row1  = cross ? (row0 + 1u) : row0;

        // Score gates: scalar cached loads, ~22x cacheline reuse across the
        // wave -> ~zero extra HBM traffic. When !cross both hit one address.
        const float cond0 = in[row0 * kC + 5u];
        const float cond1 = in[row1 * kC + 5u];

        float r[4];
        if (e0 + 4u <= n_elems) {
            const v4f x = __builtin_nontemporal_load(
                reinterpret_cast<const v4f*>(in) + v);   // global_load_b128 nt
            r[0] = x.x; r[1] = x.y; r[2] = x.z; r[3] = x.w;
        } else {
#pragma unroll
            for (int j = 0; j < 4; ++j)
                r[j] = (e0 + (unsigned int)j < n_elems) ? in[e0 + j] : 0.0f;
        }

        float o[4];
#pragma unroll
        for (int j = 0; j < 4; ++j) {
            unsigned int col  = c0 + (unsigned int)j;
            unsigned int row  = row0;
            float        cond = cond0;
            if (col >= kC) { col -= kC; row = row1; cond = cond1; }

            float val = r[j];
            // Single EXEC-masked branch: guards the anchor b128 load (must
            // not issue for every element -> would add N*85*16B of traffic)
            // and keeps v_exp_f32 off the copy lanes. Interior is branchless.
            if (col < 4u) {
                const v4f A = reinterpret_cast<const v4f*>(an)[row]; // b128
                const float mul = (col & 1u) ? A.w : A.z;            // cndmask
                const float add = (col == 0u) ? A.x
                                : ((col == 1u) ? A.y : 0.0f);        // cndmask
                const float t   = (col >= 2u) ? __expf(val) : val;   // <=4 lanes
                val = fmaf(t, mul, add);
            }
            o[j] = (cond > kThr) ? val : 0.0f;           // v_cndmask
        }

        if (e0 + 4u <= n_elems) {
            v4f y; y.x = o[0]; y.y = o[1]; y.z = o[2]; y.w = o[3];
            __builtin_nontemporal_store(y, reinterpret_cast<v4f*>(out) + v);
        } else {
#pragma unroll
            for (int j = 0; j < 4; ++j)
                if (e0 + (unsigned int)j < n_elems) out[e0 + j] = o[j];
        }
    }
}

extern "C" void kernel_launch(void* const* d_in, const int* in_sizes, int n_in,
                              void* d_out, int out_size, void* d_ws, size_t ws_size,
                              hipStream_t stream) {
    const float* in  = (const float*)d_in[0];   // [N, 85] fp32
    const float* an  = (const float*)d_in[1];   // [N, 4]  fp32
    float*       out = (float*)d_out;           // [N, 85] fp32

    const unsigned int n_elems = (unsigned int)in_sizes[0];      // N*85
    const unsigned int n_vec   = (n_elems + 3u) / 4u;

    const int          threads   = 256;
    const unsigned int per_block = (unsigned int)threads * 2u;   // 2 float4/thread
    const unsigned int blocks    = (n_vec + per_block - 1u) / per_block;

    detection_decode_kernel<<<blocks, threads, 0, stream>>>(
        in, an, out, n_elems, n_vec);
}